// Exp_loss_37168646980398
// MI455X (gfx1250) — compile-verified
//
#include <hip/hip_runtime.h>
#include <math.h>

#define C_SIZE 256
#define EPT 8              // elements per lane: 256 / wave32
#define WPB 8              // waves per block (256 threads)

typedef float v2f __attribute__((ext_vector_type(2)));
typedef float v8f __attribute__((ext_vector_type(8)));

__device__ __forceinline__ float waveSum(float v) {
#pragma unroll
    for (int m = 16; m >= 1; m >>= 1) v += __shfl_xor(v, m, 32);
    return v;
}

// Exclusive prefix over 256 elements in lane-major layout (elem i = lane*8+j).
// Within-16 part computed as a 16x16x16 f32 matmul  W = M x Uex  using four
// chained V_WMMA_F32_16X16X4_F32; the 16 row offsets come from a lane scan.
// lbase: 256-float LDS staging region private to this wave.
__device__ __forceinline__ void scan256_wmma(const float t[EPT], float p[EPT],
                                             float* lbase, int lane) {
    // stage M[r][c] = t_elem(16r+c) to LDS (lane-major write)
#pragma unroll
    for (int j = 0; j < EPT; ++j) lbase[lane * EPT + j] = t[j];
    __syncthreads();

    const int m  = lane & 15;     // A-row / D-column index
    const int hi = lane >> 4;     // half-wave select
    v8f acc = {0.f, 0.f, 0.f, 0.f, 0.f, 0.f, 0.f, 0.f};
#pragma unroll
    for (int kk = 0; kk < 4; ++kk) {
        const int k0 = 4 * kk + (hi ? 2 : 0);
        v2f a, b;
        // A = M chunk: a.x = M[m][k0], a.y = M[m][k0+1]
        a.x = lbase[16 * m + k0];
        a.y = lbase[16 * m + k0 + 1];
        // B = Uex chunk: Uex[r][c] = (r < c), column index = m
        b.x = (k0 < m) ? 1.f : 0.f;
        b.y = (k0 + 1 < m) ? 1.f : 0.f;
        acc = __builtin_amdgcn_wmma_f32_16x16x4_f32(false, a, false, b,
                                                    (short)0, acc, false, false);
    }
    __syncthreads();
    // write D back in C/D layout: VGPR g holds W[g + 8*hi][m]
#pragma unroll
    for (int g = 0; g < 8; ++g) lbase[16 * (g + 8 * hi) + m] = acc[g];
    __syncthreads();
#pragma unroll
    for (int j = 0; j < EPT; ++j) p[j] = lbase[lane * EPT + j];
    __syncthreads();

    // row offsets: offsets[r] = sum of lane totals over lanes < 2r  (r = lane/2)
    float lt = 0.f;
#pragma unroll
    for (int j = 0; j < EPT; ++j) lt += t[j];
    float ip = lt;
#pragma unroll
    for (int off = 1; off < 32; off <<= 1) {
        float v = __shfl_up(ip, off, 32);
        if (lane >= off) ip += v;
    }
    float ep   = ip - lt;                 // sum over lanes < lane
    float prev = __shfl_up(lt, 1, 32);    // lane-1 total
    float rowoff = ep - ((lane & 1) ? prev : 0.f);
#pragma unroll
    for (int j = 0; j < EPT; ++j) p[j] += rowoff;
}

__global__ void __launch_bounds__(256)
exp_loss_main(const float* __restrict__ x, const float* __restrict__ y,
              const float* __restrict__ uptr, float* __restrict__ partials,
              int B) {
    __shared__ float smem[WPB * C_SIZE];
    __shared__ float wpart[WPB];

    const int lane = threadIdx.x & 31;
    const int wid  = threadIdx.x >> 5;
    const int row  = blockIdx.x * WPB + wid;
    const bool valid = row < B;
    const int lrow = valid ? row : 0;
    const float u = uptr[0];
    float* lbase = smem + wid * C_SIZE;

    // ---- load: 8 contiguous floats per lane (2 x b128, wave covers 1KB) ----
    float xv[EPT], yv[EPT];
    {
        const float4* xb = reinterpret_cast<const float4*>(x + (size_t)lrow * C_SIZE) + lane * 2;
        const float4* yb = reinterpret_cast<const float4*>(y + (size_t)lrow * C_SIZE) + lane * 2;
        float4 x0 = xb[0], x1 = xb[1], y0 = yb[0], y1 = yb[1];
        xv[0]=x0.x; xv[1]=x0.y; xv[2]=x0.z; xv[3]=x0.w;
        xv[4]=x1.x; xv[5]=x1.y; xv[6]=x1.z; xv[7]=x1.w;
        yv[0]=y0.x; yv[1]=y0.y; yv[2]=y0.z; yv[3]=y0.w;
        yv[4]=y1.x; yv[5]=y1.y; yv[6]=y1.z; yv[7]=y1.w;
    }

    // ---- pre-sort row statistics ----
    float sy = 0.f, sxy = 0.f, su = 0.f, cntf = 0.f;
#pragma unroll
    for (int j = 0; j < EPT; ++j) {
        sy   += yv[j];
        sxy  += xv[j] * yv[j];
        su   += fmaxf(xv[j] - u, 0.f);
        cntf += (xv[j] <= u) ? 1.f : 0.f;
    }
    const float sum_y  = waveSum(sy);
    const float chosen = waveSum(sxy);
    const float S_u    = waveSum(su);
    const int   num    = (int)(waveSum(cntf) + 0.5f);
    float sch = 0.f;
#pragma unroll
    for (int j = 0; j < EPT; ++j) sch += fmaxf(xv[j] - chosen, 0.f);
    const float S_ch = waveSum(sch);
    const float A  = expf(-S_ch);
    const float Bu = expf(-S_u);
    const bool has_pos = sum_y > 0.f;

    // ---- bitonic sort of 256 (x,y) pairs, index i = lane*8 + j, ascending ----
#pragma unroll
    for (int k = 2; k <= 256; k <<= 1) {
#pragma unroll
        for (int d = 128; d > 0; d >>= 1) {
            if (d >= k) continue;
            if (d >= 8) {
                const int  ld      = d >> 3;
                const bool up      = (((lane << 3) & k) == 0);
                const bool lower   = ((lane & ld) == 0);
                const bool takeMin = (up == lower);
#pragma unroll
                for (int j = 0; j < EPT; ++j) {
                    float ox = __shfl_xor(xv[j], ld, 32);
                    float oy = __shfl_xor(yv[j], ld, 32);
                    bool own = takeMin ? (xv[j] <= ox) : (xv[j] > ox);
                    xv[j] = own ? xv[j] : ox;
                    yv[j] = own ? yv[j] : oy;
                }
            } else {
#pragma unroll
                for (int j = 0; j < EPT; ++j) {
                    if ((j & d) == 0) {
                        const int  pj = j | d;
                        const bool up = ((((lane << 3) | j) & k) == 0);
                        bool sw = up ? (xv[j] > xv[pj]) : (xv[j] < xv[pj]);
                        if (sw) {
                            float tx = xv[j]; xv[j] = xv[pj]; xv[pj] = tx;
                            float ty = yv[j]; yv[j] = yv[pj]; yv[pj] = ty;
                        }
                    }
                }
            }
        }
    }

    // ---- suffix sums -> S_i, E, F, per-element scan inputs ----
    float lts = 0.f;
#pragma unroll
    for (int j = 0; j < EPT; ++j) lts += xv[j];
    float is = lts;  // inclusive suffix over lanes
#pragma unroll
    for (int off = 1; off < 32; off <<= 1) {
        float v = __shfl_down(is, off, 32);
        if (lane + off < 32) is += v;
    }

    float eterm[EPT], fterm[EPT], exu[EPT];
    float run = 0.f, mf = 0.f, me = 0.f;
#pragma unroll
    for (int j = 0; j < EPT; ++j) {
        const int i = lane * EPT + j;
        const float xs = xv[j];
        const float T  = is - run;            // sum over m >= i of xs_m
        run += xs;
        const float S  = T - (float)(256 - i) * xs;
        const float E  = expf(-S);
        exu[j] = expf(xs - u);
        const float F  = E * exu[j];
        fterm[j] = F / (float)((256 - i) * (257 - i));
        eterm[j] = (i < 255) ? E / (float)((255 - i) * (256 - i)) : 0.f;
        if (i < num) { mf += fterm[j]; me += eterm[j]; }
    }
    const float cumF_num = waveSum(mf);
    const float cumE_num = waveSum(me);

    // ---- exclusive prefix scans via WMMA ----
    float pF[EPT], pE[EPT];
    scan256_wmma(fterm, pF, lbase, lane);
    scan256_wmma(eterm, pE, lbase, lane);

    // ---- per-element terms, row value ----
    const float bu_den = (num < 256) ? (float)((256 - num) * (257 - num)) : 1.f;
    float accp = 0.f;
#pragma unroll
    for (int j = 0; j < EPT; ++j) {
        const int i = lane * EPT + j;
        float term;
        if (i < num)
            term = A * exu[j] / (float)(257 - i) - pF[j];
        else
            term = A / (float)(256 - i) - Bu / bu_den - cumF_num - (pE[j] - cumE_num);
        accp += (yv[j] != 0.f) ? term : 0.f;
    }
    const float rp = waveSum(accp);
    const float row_val = has_pos ? rp : (Bu / (float)(257 - num) - cumF_num);

    if (lane == 0) wpart[wid] = valid ? row_val : 0.f;
    __syncthreads();
    if (threadIdx.x == 0) {
        float s = 0.f;
#pragma unroll
        for (int w = 0; w < WPB; ++w) s += wpart[w];
        partials[blockIdx.x] = s;
    }
}

__global__ void __launch_bounds__(256)
exp_loss_reduce(const float* __restrict__ partials, int n,
                float* __restrict__ out, float invB) {
    __shared__ float s[256];
    float v = 0.f;
    for (int i = threadIdx.x; i < n; i += 256) v += partials[i];
    s[threadIdx.x] = v;
    __syncthreads();
    for (int k = 128; k > 0; k >>= 1) {
        if ((int)threadIdx.x < k) s[threadIdx.x] += s[threadIdx.x + k];
        __syncthreads();
    }
    if (threadIdx.x == 0) out[0] = -s[0] * invB;
}

extern "C" void kernel_launch(void* const* d_in, const int* in_sizes, int n_in,
                              void* d_out, int out_size, void* d_ws, size_t ws_size,
                              hipStream_t stream) {
    const float* x = (const float*)d_in[0];
    const float* y = (const float*)d_in[1];
    const float* u = (const float*)d_in[2];
    float* out = (float*)d_out;
    const int B = in_sizes[0] / C_SIZE;
    const int nblocks = (B + WPB - 1) / WPB;
    float* partials = (float*)d_ws;   // nblocks floats of scratch

    exp_loss_main<<<nblocks, 32 * WPB, 0, stream>>>(x, y, u, partials, B);
    exp_loss_reduce<<<1, 256, 0, stream>>>(partials, nblocks, out, 1.0f / (float)B);
}